// VectorQuantizer_2130303779188
// MI455X (gfx1250) — compile-verified
//
#include <hip/hip_runtime.h>
#include <hip/hip_bf16.h>

typedef float v2f __attribute__((ext_vector_type(2)));
typedef float v8f __attribute__((ext_vector_type(8)));

#define BB      32
#define CC      64
#define KK      2048
#define NTOT    (BB * KK)          // 65536 rows
#define NCODES  1024
#define NELEM   (BB * CC * KK)     // 4194304 z elements
#define MTILES  2                  // 16-row tiles per wave (B-fragment reuse)
#define ROWS_PER_WAVE (16 * MTILES)
#define WAVES_PER_WG 8
#define MAIN_BLOCKS  (NTOT / (ROWS_PER_WAVE * WAVES_PER_WG))   // 256

// workspace layout (4-byte words)
#define WS_HIST    0      // 1024 u32
#define WS_PART    1024   // MAIN_BLOCKS f32 (per-WG loss partials)
#define WS_ESQ     2048   // 1024 f32 (||e||^2)
#define WS_WORDS   3072

__global__ __launch_bounds__(256) void vq_zero(unsigned int* ws) {
    int i = blockIdx.x * blockDim.x + threadIdx.x;
    if (i < WS_PART + MAIN_BLOCKS) ws[i] = 0u;   // zero hist + partials
}

__global__ __launch_bounds__(256) void vq_esq(const float* __restrict__ cb,
                                              float* __restrict__ eSq) {
    int j = blockIdx.x * blockDim.x + threadIdx.x;
    if (j < NCODES) {
        const float4* p = (const float4*)(cb + j * CC);
        float s = 0.f;
#pragma unroll
        for (int i = 0; i < CC / 4; ++i) {
            float4 v = p[i];
            s += v.x * v.x + v.y * v.y + v.z * v.z + v.w * v.w;
        }
        eSq[j] = s;
    }
}

__global__ __launch_bounds__(256) void vq_main(const float* __restrict__ z_e,
                                               const float* __restrict__ cb,
                                               const float* __restrict__ eSq,
                                               float* __restrict__ zq_out,
                                               float* __restrict__ codes_out,
                                               unsigned int* __restrict__ hist,
                                               float* __restrict__ wg_partial) {
    const int lane     = threadIdx.x & 31;
    const int waveInWg = threadIdx.x >> 5;
    const int waveId   = blockIdx.x * WAVES_PER_WG + waveInWg;
    const int rowBase  = waveId * ROWS_PER_WAVE;   // 32 rows per wave
    const int b        = rowBase >> 11;            // rowBase / K
    const int kbase    = rowBase & (KK - 1);       // 32-aligned => same b for all rows
    const int m        = lane & 15;                // row within 16-tile / col (B,C)
    const int kOff     = (lane >> 4) * 2;          // K sub-offset per wmma fragment

    // ---- load A fragments: MTILES x 16 rows of z stay in registers ----
    v2f a[MTILES][16];
#pragma unroll
    for (int mt = 0; mt < MTILES; ++mt) {
        const float* zb = z_e + (size_t)b * (CC * KK) + kbase + mt * 16 + m;
#pragma unroll
        for (int s = 0; s < 16; ++s) {
            int c0 = 4 * s + kOff;
            a[mt][s].x = zb[(size_t)c0 * KK];
            a[mt][s].y = zb[(size_t)(c0 + 1) * KK];
        }
    }

    float best[MTILES][8];
    int   bidx[MTILES][8];
#pragma unroll
    for (int mt = 0; mt < MTILES; ++mt)
#pragma unroll
        for (int r = 0; r < 8; ++r) { best[mt][r] = 3.402823466e38f; bidx[mt][r] = 0; }

    // ---- sweep all 1024 codes in tiles of 16; each B fragment feeds MTILES WMMAs ----
    for (int t = 0; t < NCODES / 16; ++t) {
        const int jl = t * 16 + m;                 // this lane's code column
        const float* cbb = cb + jl * CC + kOff;
        v8f acc[MTILES];
#pragma unroll
        for (int mt = 0; mt < MTILES; ++mt)
            acc[mt] = (v8f){0.f, 0.f, 0.f, 0.f, 0.f, 0.f, 0.f, 0.f};
#pragma unroll
        for (int s = 0; s < 16; ++s) {
            float2 tb = *(const float2*)(cbb + 4 * s);
            v2f bf; bf.x = tb.x; bf.y = tb.y;
#pragma unroll
            for (int mt = 0; mt < MTILES; ++mt)
                acc[mt] = __builtin_amdgcn_wmma_f32_16x16x4_f32(
                    false, a[mt][s], false, bf, (short)0, acc[mt], false, false);
        }
        const float es = eSq[jl];
#pragma unroll
        for (int mt = 0; mt < MTILES; ++mt)
#pragma unroll
            for (int r = 0; r < 8; ++r) {
                float sc = fmaf(-2.0f, acc[mt][r], es);   // ||e||^2 - 2 z.e
                if (sc < best[mt][r]) { best[mt][r] = sc; bidx[mt][r] = jl; }
            }
    }

    // ---- argmin across the 16 lanes of each half (columns), tie -> lower idx ----
#pragma unroll
    for (int mt = 0; mt < MTILES; ++mt)
#pragma unroll
        for (int r = 0; r < 8; ++r)
#pragma unroll
            for (int mask = 1; mask < 16; mask <<= 1) {
                float os = __shfl_xor(best[mt][r], mask, 32);
                int   oi = __shfl_xor(bidx[mt][r], mask, 32);
                if (os < best[mt][r] || (os == best[mt][r] && oi < bidx[mt][r])) {
                    best[mt][r] = os; bidx[mt][r] = oi;
                }
            }
    // lanes 0-15 agree on rows 0..7 (reg r), lanes 16-31 on rows 8..15 (per tile)

#pragma unroll
    for (int mt = 0; mt < MTILES; ++mt) {
        // compile-time-indexed select of this lane's writer value
        int codeSel = bidx[mt][0];
#pragma unroll
        for (int r = 1; r < 8; ++r)
            if ((lane & 7) == r) codeSel = bidx[mt][r];

        if ((lane & 15) < 8) {                     // 16 writer lanes: rows 0..15
            int row = ((lane >> 4) << 3) + (lane & 7);
            codes_out[rowBase + mt * 16 + row] = (float)codeSel;
            atomicAdd(&hist[codeSel], 1u);
        }
    }

    // broadcast: lane needs the code of row (lane & 15) in each tile
    int myCode[MTILES];
#pragma unroll
    for (int mt = 0; mt < MTILES; ++mt) {
        myCode[mt] = 0;
#pragma unroll
        for (int r = 0; r < 8; ++r) {
            int lo = __shfl(bidx[mt][r], 0, 32);   // row r
            int hi = __shfl(bidx[mt][r], 16, 32);  // row 8 + r
            if ((lane & 7) == r) myCode[mt] = (lane & 8) ? hi : lo;
        }
    }

    // ---- fused z_q scatter + (z_q - z_e)^2 loss (A regs still hold z_e) ----
    float lacc = 0.f;
#pragma unroll
    for (int mt = 0; mt < MTILES; ++mt) {
        const float* qb = cb + myCode[mt] * CC + kOff;
        float* ob = zq_out + (size_t)b * (CC * KK) + kbase + mt * 16 + m;
#pragma unroll
        for (int s = 0; s < 16; ++s) {
            int c0 = 4 * s + kOff;
            float2 q = *(const float2*)(qb + 4 * s);
            ob[(size_t)c0 * KK]       = q.x;
            ob[(size_t)(c0 + 1) * KK] = q.y;
            float d0 = a[mt][s].x - q.x, d1 = a[mt][s].y - q.y;
            lacc = fmaf(d0, d0, lacc);
            lacc = fmaf(d1, d1, lacc);
        }
    }
    // deterministic loss reduction: wave -> LDS -> per-WG partial
#pragma unroll
    for (int mask = 16; mask >= 1; mask >>= 1)
        lacc += __shfl_xor(lacc, mask, 32);
    __shared__ float sp[WAVES_PER_WG];
    if (lane == 0) sp[waveInWg] = lacc;
    __syncthreads();
    if (threadIdx.x == 0) {
        float s = 0.f;
#pragma unroll
        for (int i = 0; i < WAVES_PER_WG; ++i) s += sp[i];
        wg_partial[blockIdx.x] = s;
    }
}

__global__ __launch_bounds__(1024) void vq_finalize(const unsigned int* __restrict__ hist,
                                                    const float* __restrict__ wg_partial,
                                                    float* __restrict__ out_loss,
                                                    float* __restrict__ out_perp) {
    __shared__ float red[1024];
    const int tid = threadIdx.x;

    // entropy over histogram
    float p = (float)hist[tid] * (1.0f / (float)NTOT);
    red[tid] = p * logf(p + 1e-10f);
    __syncthreads();
    for (int s = 512; s > 0; s >>= 1) {
        if (tid < s) red[tid] += red[tid + s];
        __syncthreads();
    }
    float entSum = red[0];
    __syncthreads();

    // loss partials (fixed-order tree => deterministic)
    red[tid] = (tid < MAIN_BLOCKS) ? wg_partial[tid] : 0.f;
    __syncthreads();
    for (int s = 512; s > 0; s >>= 1) {
        if (tid < s) red[tid] += red[tid + s];
        __syncthreads();
    }
    if (tid == 0) {
        *out_loss = 1.25f * red[0] / (float)NELEM;  // (1 + beta) * mse
        *out_perp = expf(-entSum);
    }
}

extern "C" void kernel_launch(void* const* d_in, const int* in_sizes, int n_in,
                              void* d_out, int out_size, void* d_ws, size_t ws_size,
                              hipStream_t stream) {
    const float* z_e = (const float*)d_in[0];
    const float* cb  = (const float*)d_in[1];

    float* out       = (float*)d_out;
    float* zq_out    = out;                     // NELEM
    float* codes_out = out + NELEM;             // NTOT
    float* loss_out  = out + NELEM + NTOT;      // 1
    float* perp_out  = out + NELEM + NTOT + 1;  // 1

    unsigned int* ws_u = (unsigned int*)d_ws;
    unsigned int* hist = ws_u + WS_HIST;
    float* partials    = (float*)(ws_u + WS_PART);
    float* eSq         = (float*)(ws_u + WS_ESQ);

    vq_zero<<<(WS_PART + MAIN_BLOCKS + 255) / 256, 256, 0, stream>>>(ws_u);
    vq_esq<<<(NCODES + 255) / 256, 256, 0, stream>>>(cb, eSq);
    vq_main<<<MAIN_BLOCKS, 32 * WAVES_PER_WG, 0, stream>>>(
        z_e, cb, eSq, zq_out, codes_out, hist, partials);
    vq_finalize<<<1, 1024, 0, stream>>>(hist, partials, loss_out, perp_out);
}